// CLING_HAN_16406775071378
// MI455X (gfx1250) — compile-verified
//
#include <hip/hip_runtime.h>
#include <cstdint>

// ---------------- problem constants ----------------
#define HAN_NODES   200000
#define HAN_DEG     32
#define HAN_FEAT    128

typedef float v2f __attribute__((ext_vector_type(2)));
typedef float v8f __attribute__((ext_vector_type(8)));

#define WMMA_F32(a, b, c) \
  __builtin_amdgcn_wmma_f32_16x16x4_f32(false, (a), false, (b), (short)0, (c), false, false)

// -------- JAX Threefry-2x32 (20 rounds), host + device --------
__host__ __device__ inline void threefry2x32(uint32_t k0, uint32_t k1,
                                             uint32_t x0, uint32_t x1,
                                             uint32_t* o0, uint32_t* o1) {
  const uint32_t ks2 = k0 ^ k1 ^ 0x1BD11BDAu;
  uint32_t y0 = x0 + k0, y1 = x1 + k1;
#define TF_ROT(r) { y0 += y1; y1 = (y1 << (r)) | (y1 >> (32 - (r))); y1 ^= y0; }
  TF_ROT(13) TF_ROT(15) TF_ROT(26) TF_ROT(6)
  y0 += k1;  y1 += ks2 + 1u;
  TF_ROT(17) TF_ROT(29) TF_ROT(16) TF_ROT(24)
  y0 += ks2; y1 += k0 + 2u;
  TF_ROT(13) TF_ROT(15) TF_ROT(26) TF_ROT(6)
  y0 += k0;  y1 += k1 + 3u;
  TF_ROT(17) TF_ROT(29) TF_ROT(16) TF_ROT(24)
  y0 += k1;  y1 += ks2 + 4u;
  TF_ROT(13) TF_ROT(15) TF_ROT(26) TF_ROT(6)
  y0 += ks2; y1 += k0 + 5u;
#undef TF_ROT
  *o0 = y0; *o1 = y1;
}

// -------- neighbor sampling: next[i] = adj[prev[i/16]][bits(i)&31] --------
__global__ void __launch_bounds__(256) han_sample(const int* __restrict__ prev,
                                                  const int* __restrict__ adj,
                                                  int* __restrict__ next,
                                                  uint32_t k0, uint32_t k1, int n) {
  int i = blockIdx.x * blockDim.x + threadIdx.x;
  if (i >= n) return;
  const int half = n >> 1;                 // n even (8192 / 131072)
  uint32_t j = (i < half) ? (uint32_t)i : (uint32_t)(i - half);
  uint32_t o0, o1;
  threefry2x32(k0, k1, j, j + (uint32_t)half, &o0, &o1);
  uint32_t bits = (i < half) ? o0 : o1;    // randint span 32 -> bits & 31
  int col  = (int)(bits & 31u);
  int node = prev[i >> 4];
  next[i]  = adj[(size_t)node * HAN_DEG + col];
}

// -------- repack [H][K][128] weight into WMMA-B fragment order --------
// dest (per head): [nt:8][k2:K/4][lane:32][j:2], value = W[4*k2 + 2*(lane>>4) + j][nt*16 + (lane&15)]
__global__ void __launch_bounds__(256) han_repack(const float* __restrict__ W,
                                                  float* __restrict__ Wo, int K, int H) {
  int i = blockIdx.x * 256 + threadIdx.x;
  int per = K * 128;
  if (i >= H * per) return;
  int h = i / per, rem = i - h * per;
  int l2 = rem & 63, t2 = rem >> 6;
  int k4 = K >> 2;
  int k2 = t2 % k4, nt = t2 / k4;
  int j = l2 & 1, lanex = l2 >> 1, g = lanex >> 4, col = lanex & 15;
  Wo[i] = W[(size_t)h * per + (4 * k2 + 2 * g + j) * 128 + nt * 16 + col];
}

// -------- prep: out[m] = feats[idx[m]] @ Wp (repacked), 16 rows/block --------
__global__ void __launch_bounds__(256) han_prep(const int* __restrict__ idx,
                                                const float* __restrict__ feats,
                                                const float* __restrict__ Wp2,
                                                float* __restrict__ out) {
  __shared__ float At[16 * 132];
  const int t  = threadIdx.x;
  const int m0 = blockIdx.x * 16;
  for (int e = t; e < 16 * 32; e += 256) {
    int r = e >> 5, c4 = e & 31;
    int node = idx[m0 + r];
    float4 v = ((const float4*)(feats + (size_t)node * HAN_FEAT))[c4];
    *((float4*)&At[r * 132 + c4 * 4]) = v;
  }
  __syncthreads();
  const int lane = t & 31, nt = t >> 5;
  const int row = lane & 15, koff = (lane >> 4) * 2, col = lane & 15;
  const int rbase = (lane >> 4) * 8;
  v8f c = {0, 0, 0, 0, 0, 0, 0, 0};
  for (int k2 = 0; k2 < 32; k2++) {
    v2f a = *(const v2f*)&At[row * 132 + 4 * k2 + koff];
    v2f b = *(const v2f*)(Wp2 + ((size_t)(nt * 32 + k2)) * 64 + lane * 2);
    c = WMMA_F32(a, b, c);
  }
#pragma unroll
  for (int i = 0; i < 8; i++)
    out[(size_t)(m0 + rbase + i) * 128 + nt * 16 + col] = c[i];
}

// -------- one attention-agg layer; 2 output rows per block --------
// prep==1 (d==128): nb = feats[cur[...]] @ Wp2 computed in-block.
// prep==0: nb loaded from nbbuf.
__global__ void __launch_bounds__(256) han_agg(
    const float* __restrict__ xbuf,   // [M, d]
    const float* __restrict__ nbbuf,  // [M*16, d] or null
    const int*   __restrict__ cur,    // [M*16] or null
    const float* __restrict__ feats,
    const float* __restrict__ Wp2,    // repacked prep weight
    const float* __restrict__ Wa1p,   // repacked: [4][8][K2/4][64]
    const float* __restrict__ wa2,    // [4, 128]
    const float* __restrict__ Wx,     // [4, d, 64]
    const float* __restrict__ Wn,     // [4, d, 64]
    float* __restrict__ out,          // [M, 256]
    int d, int prep) {
  __shared__ float s_At[16 * 132];        // prep gather tile (reused per row)
  __shared__ float s_x [2][256];
  __shared__ float s_nb[2][16 * 260];     // stride 260: 16B aligned, conflict-free
  __shared__ float s_ps[2][8 * 4 * 16];   // per-nt score partials [nt][h][s]
  __shared__ float s_sc[2][64];
  __shared__ float s_al[2][64];
  __shared__ float s_ag[2][4 * 256];

  const int t = threadIdx.x;
  const int m0 = blockIdx.x * 2;
  const int lane = t & 31, wave = t >> 5;  // wave == N-tile nt
  const int row = lane & 15, koff = (lane >> 4) * 2, col = lane & 15;
  const int rbase = (lane >> 4) * 8;
  const int dshift = (d == 256) ? 8 : 7;
  const int d4 = d >> 2;                   // k2 steps per half
  const int kstride = d >> 1;              // K2/4

  for (int e = t; e < 2 * d; e += 256) {
    int r = e >> dshift, dd = e & (d - 1);
    s_x[r][dd] = xbuf[(size_t)(m0 + r) * d + dd];
  }

  if (prep) {  // uniform branch
    for (int r = 0; r < 2; r++) {
      for (int e = t; e < 16 * 32; e += 256) {
        int rr = e >> 5, c4 = e & 31;
        int node = cur[(m0 + r) * 16 + rr];
        float4 v = ((const float4*)(feats + (size_t)node * HAN_FEAT))[c4];
        *((float4*)&s_At[rr * 132 + c4 * 4]) = v;
      }
      __syncthreads();
      v8f c = {0, 0, 0, 0, 0, 0, 0, 0};
      for (int k2 = 0; k2 < 32; k2++) {
        v2f a = *(const v2f*)&s_At[row * 132 + 4 * k2 + koff];
        v2f b = *(const v2f*)(Wp2 + ((size_t)(wave * 32 + k2)) * 64 + lane * 2);
        c = WMMA_F32(a, b, c);
      }
#pragma unroll
      for (int i = 0; i < 8; i++)
        s_nb[r][(rbase + i) * 260 + wave * 16 + col] = c[i];
      __syncthreads();   // protect s_At reuse + publish s_nb[r]
    }
  } else {
    const int per = 16 * d4;               // float4 per row
    for (int e = t; e < 2 * per; e += 256) {
      int r = e / per, rem = e - r * per;
      int rr = rem / d4, c4 = rem - rr * d4;
      float4 v = ((const float4*)(nbbuf + ((size_t)(m0 + r) * 16 + rr) * d))[c4];
      *((float4*)&s_nb[r][rr * 260 + c4 * 4]) = v;
    }
  }
  __syncthreads();

  // h = tanh(cat @ Wa1[h]); wave owns nt=wave for all 4 heads x 2 rows (8 accums)
  v8f acc[2][4];
  {
    v8f z = {0, 0, 0, 0, 0, 0, 0, 0};
#pragma unroll
    for (int r = 0; r < 2; r++)
#pragma unroll
      for (int h = 0; h < 4; h++) acc[r][h] = z;
  }
  // k in [0,d): cat column = x (identical across the 16 sample rows)
  for (int k2 = 0; k2 < d4; k2++) {
    int ka = 4 * k2 + koff;
    v2f a0 = *(const v2f*)&s_x[0][ka];
    v2f a1 = *(const v2f*)&s_x[1][ka];
#pragma unroll
    for (int h = 0; h < 4; h++) {
      v2f b = *(const v2f*)(Wa1p + (((size_t)(h * 8 + wave)) * kstride + k2) * 64 + lane * 2);
      acc[0][h] = WMMA_F32(a0, b, acc[0][h]);
      acc[1][h] = WMMA_F32(a1, b, acc[1][h]);
    }
  }
  // k in [d,2d): cat column = nb[s]
  for (int k2 = 0; k2 < d4; k2++) {
    int ka = 4 * k2 + koff;
    v2f a0 = *(const v2f*)&s_nb[0][row * 260 + ka];
    v2f a1 = *(const v2f*)&s_nb[1][row * 260 + ka];
#pragma unroll
    for (int h = 0; h < 4; h++) {
      v2f b = *(const v2f*)(Wa1p + (((size_t)(h * 8 + wave)) * kstride + (k2 + d4)) * 64 + lane * 2);
      acc[0][h] = WMMA_F32(a0, b, acc[0][h]);
      acc[1][h] = WMMA_F32(a1, b, acc[1][h]);
    }
  }

  // score partials: sum over this tile's 16 columns via deterministic xor-butterfly
#pragma unroll
  for (int h = 0; h < 4; h++) {
    float w2 = wa2[h * 128 + wave * 16 + col];
#pragma unroll
    for (int r = 0; r < 2; r++) {
#pragma unroll
      for (int i = 0; i < 8; i++) {
        float v = tanhf(acc[r][h][i]) * w2;
        v += __shfl_xor(v, 1);
        v += __shfl_xor(v, 2);
        v += __shfl_xor(v, 4);
        v += __shfl_xor(v, 8);
        if (col == 0) s_ps[r][(wave * 4 + h) * 16 + rbase + i] = v;
      }
    }
  }
  __syncthreads();

  // score[h][s] = sum over the 8 nt partials (fixed order -> deterministic)
  if (t < 128) {
    int r = t >> 6, h = (t >> 4) & 3, s = t & 15;
    float a = 0.f;
#pragma unroll
    for (int nt = 0; nt < 8; nt++) a += s_ps[r][(nt * 4 + h) * 16 + s];
    s_sc[r][h * 16 + s] = a;
  }
  __syncthreads();

  // softmax over S=16 per (row, head)
  if (t < 8) {
    int r = t >> 2, h = t & 3;
    float mx = -1e30f;
    for (int s = 0; s < 16; s++) mx = fmaxf(mx, s_sc[r][h * 16 + s]);
    float e[16], sum = 0.f;
    for (int s = 0; s < 16; s++) { e[s] = expf(s_sc[r][h * 16 + s] - mx); sum += e[s]; }
    float inv = 1.f / sum;
    for (int s = 0; s < 16; s++) s_al[r][h * 16 + s] = e[s] * inv;
  }
  __syncthreads();

  // agg[h][dd] = sum_s alpha[h][s] * nb[s][dd]
  for (int e2 = t; e2 < 2 * 4 * d; e2 += 256) {
    int r = e2 >> (dshift + 2), rem = e2 & (4 * d - 1);
    int h = rem >> dshift, dd = rem & (d - 1);
    float a = 0.f;
#pragma unroll
    for (int s = 0; s < 16; s++) a += s_al[r][h * 16 + s] * s_nb[r][s * 260 + dd];
    s_ag[r][h * 256 + dd] = a;
  }
  __syncthreads();

  // out = relu(x . Wx + agg . Wn); each weight load feeds both rows
  {
    int h = t >> 6, o = t & 63;
    const float* wx = Wx + (size_t)h * d * 64;
    const float* wn = Wn + (size_t)h * d * 64;
    float a0 = 0.f, a1 = 0.f;
    for (int dd = 0; dd < d; dd++) {
      float wxv = wx[dd * 64 + o], wnv = wn[dd * 64 + o];
      a0 += s_x[0][dd] * wxv + s_ag[0][h * 256 + dd] * wnv;
      a1 += s_x[1][dd] * wxv + s_ag[1][h * 256 + dd] * wnv;
    }
    out[(size_t)m0 * 256 + t]       = fmaxf(a0, 0.f);
    out[(size_t)(m0 + 1) * 256 + t] = fmaxf(a1, 0.f);
  }
}

extern "C" void kernel_launch(void* const* d_in, const int* in_sizes, int n_in,
                              void* d_out, int out_size, void* d_ws, size_t ws_size,
                              hipStream_t stream) {
  const int*   ids   = (const int*)  d_in[0];
  const float* feats = (const float*)d_in[1];
  const int*   adjs  = (const int*)  d_in[2];
  const float* Wp    = (const float*)d_in[3];
  const float* Wa1_0 = (const float*)d_in[4];
  const float* wa2_0 = (const float*)d_in[5];
  const float* Wx_0  = (const float*)d_in[6];
  const float* Wn_0  = (const float*)d_in[7];
  const float* Wa1_1 = (const float*)d_in[8];
  const float* wa2_1 = (const float*)d_in[9];
  const float* Wx_1  = (const float*)d_in[10];
  const float* Wn_1  = (const float*)d_in[11];
  float* outp = (float*)d_out;

  // workspace layout (~15.6 MB)
  int*   cur1  = (int*)d_ws;                // 8192
  int*   cur2  = cur1 + 8192;               // 131072
  float* P0    = (float*)(cur2 + 131072);   // 512*128
  float* P1    = P0 + 512 * 128;            // 8192*128
  float* o0    = P1 + 8192 * 128;           // 512*256
  float* o1    = o0 + 512 * 256;            // 8192*256
  float* Wp2   = o1 + 8192 * 256;           // 128*128   (repacked)
  float* Wa1p0 = Wp2 + 128 * 128;           // 4*256*128 (repacked, per-mp)
  float* Wa1p1 = Wa1p0 + 4 * 256 * 128;     // 4*512*128 (repacked, per-mp)

  // fold_in(key(42), mp*16+li): key(42) = (0,42); counts = (0, data)
  uint32_t key[2][2][2];
  for (int mp = 0; mp < 2; mp++)
    for (int li = 0; li < 2; li++)
      threefry2x32(0u, 42u, 0u, (uint32_t)(mp * 16 + li),
                   &key[mp][li][0], &key[mp][li][1]);

  // repack prep weight once; init = feats[ids] @ Wp
  han_repack<<<(128 * 128 + 255) / 256, 256, 0, stream>>>(Wp, Wp2, 128, 1);
  han_prep<<<512 / 16, 256, 0, stream>>>(ids, feats, Wp2, P0);

  for (int mp = 0; mp < 2; mp++) {
    const int* adj = adjs + (size_t)mp * HAN_NODES * HAN_DEG;
    han_sample<<<8192 / 256,   256, 0, stream>>>(ids,  adj, cur1, key[mp][0][0], key[mp][0][1], 8192);
    han_sample<<<131072 / 256, 256, 0, stream>>>(cur1, adj, cur2, key[mp][1][0], key[mp][1][1], 131072);
    han_prep<<<8192 / 16, 256, 0, stream>>>(cur1, feats, Wp2, P1);

    // repack this mp's attention weights into WMMA-B fragment order
    han_repack<<<(4 * 256 * 128 + 255) / 256, 256, 0, stream>>>(
        Wa1_0 + (size_t)mp * 4 * 256 * 128, Wa1p0, 256, 4);
    han_repack<<<(4 * 512 * 128 + 255) / 256, 256, 0, stream>>>(
        Wa1_1 + (size_t)mp * 4 * 512 * 128, Wa1p1, 512, 4);

    const float* a2 = wa2_0 + (size_t)mp * 4 * 128;
    const float* ax = Wx_0  + (size_t)mp * 4 * 128 * 64;
    const float* an = Wn_0  + (size_t)mp * 4 * 128 * 64;
    // round 0: k_=1 (fused level-2 prep), then k_=0
    han_agg<<<8192 / 2, 256, 0, stream>>>(P1, nullptr, cur2, feats, Wp2, Wa1p0, a2, ax, an, o1, 128, 1);
    han_agg<<<512 / 2,  256, 0, stream>>>(P0, P1, nullptr, feats, Wp2, Wa1p0, a2, ax, an, o0, 128, 0);

    const float* b2 = wa2_1 + (size_t)mp * 4 * 128;
    const float* bx = Wx_1  + (size_t)mp * 4 * 256 * 64;
    const float* bn = Wn_1  + (size_t)mp * 4 * 256 * 64;
    // round 1 -> final output slice [mp, 512, 256]
    han_agg<<<512 / 2, 256, 0, stream>>>(o0, o1, nullptr, feats, Wp2, Wa1p1, b2, bx, bn,
                                         outp + (size_t)mp * 512 * 256, 256, 0);
  }
}